// my_SimCLR_7241314861695
// MI455X (gfx1250) — compile-verified
//
#include <hip/hip_runtime.h>

typedef float v2f __attribute__((ext_vector_type(2)));
typedef float v8f __attribute__((ext_vector_type(8)));

#define NB    32      // batch (one workgroup per sample)
#define NK    16      // pos+neg samples per query
#define D_IN  312
#define H1D   1024
#define H2D   2048
#define D_OUT 128
#define H2_CHUNK 256

// LDS strides padded for conflict-free WMMA A-operand reads (gcd(stride,64)==1)
#define SX_STRIDE  313
#define SH1_STRIDE 1025
#define SH2_STRIDE 257

__device__ __forceinline__ v8f wmma4(v2f a, v2f b, v8f c) {
  // V_WMMA_F32_16X16X4_F32: D = A(16x4) * B(4x16) + C(16x16)
  return __builtin_amdgcn_wmma_f32_16x16x4_f32(
      /*neg_a=*/false, a, /*neg_b=*/false, b,
      /*c_mod=*/(short)0, c, /*reuse_a=*/false, /*reuse_b=*/false);
}

__global__ __launch_bounds__(256)
void simclr_expert_mlp_kernel(
    const float* __restrict__ v2s,
    const float* __restrict__ W1, const float* __restrict__ b1,
    const float* __restrict__ W2, const float* __restrict__ b2,
    const float* __restrict__ W3, const float* __restrict__ b3,
    const int* __restrict__ tar_atts,
    const int* __restrict__ sample_idx,
    float* __restrict__ out)
{
  extern __shared__ float smem[];
  float* sX  = smem;                         // 32 x SX_STRIDE   (inputs; later reused as sOut)
  float* sH1 = sX  + 32 * SX_STRIDE;         // 32 x SH1_STRIDE  (layer-1 activations)
  float* sH2 = sH1 + 32 * SH1_STRIDE;        // 32 x SH2_STRIDE  (layer-2 chunk)

  const int b    = blockIdx.x;
  const int tid  = threadIdx.x;
  const int wave = tid >> 5;
  const int lane = tid & 31;
  const int ln16 = lane & 15;
  const int hi   = lane >> 4;                // half-wave select
  const int kofs = hi << 1;                  // K sub-offset inside a 4-wide WMMA step

  const int e = tar_atts[b];
  const float* W1e = W1 + (size_t)e * D_IN * H1D;
  const float* W2e = W2 + (size_t)e * H1D * H2D;
  const float* W3e = W3 + (size_t)e * H2D * D_OUT;
  const float* b1e = b1 + (size_t)e * H1D;
  const float* b2e = b2 + (size_t)e * H2D;
  const float* b3e = b3 + (size_t)e * D_OUT;

  // ---- Stage activations: row 0 = x[b]; rows 1..16 = x[sample_idx[b,k]]; 17..31 = 0
  for (int idx = tid; idx < 32 * D_IN; idx += 256) {
    const int r = idx / D_IN;
    const int c = idx - r * D_IN;
    float v = 0.0f;
    if (r == 0)       v = v2s[(size_t)b * D_IN + c];
    else if (r <= NK) v = v2s[(size_t)sample_idx[b * NK + (r - 1)] * D_IN + c];
    sX[r * SX_STRIDE + c] = v;
  }
  __syncthreads();

  // ---------------- Layer 1: h1 = relu(x @ W1[e] + b1[e])  [32 x 1024] ----------------
  for (int nt = wave; nt < H1D / 16; nt += 8) {
    const int ncol = nt * 16 + ln16;
    v8f acc0 = {0,0,0,0,0,0,0,0};
    v8f acc1 = {0,0,0,0,0,0,0,0};
#pragma unroll 4
    for (int k = 0; k < D_IN; k += 4) {
      const int k0 = k + kofs;
      v2f a0, a1, bb;
      a0.x = sX[ln16 * SX_STRIDE + k0];
      a0.y = sX[ln16 * SX_STRIDE + k0 + 1];
      a1.x = sX[(ln16 + 16) * SX_STRIDE + k0];
      a1.y = sX[(ln16 + 16) * SX_STRIDE + k0 + 1];
      bb.x = W1e[(size_t)k0 * H1D + ncol];
      bb.y = W1e[(size_t)(k0 + 1) * H1D + ncol];
      acc0 = wmma4(a0, bb, acc0);
      acc1 = wmma4(a1, bb, acc1);
    }
    const float bias = b1e[ncol];
#pragma unroll
    for (int r = 0; r < 8; ++r) {
      const int row = r + (hi << 3);
      sH1[row * SH1_STRIDE + ncol]        = fmaxf(acc0[r] + bias, 0.0f);
      sH1[(row + 16) * SH1_STRIDE + ncol] = fmaxf(acc1[r] + bias, 0.0f);
    }
  }
  __syncthreads();

  // ---- Layers 2+3 fused over H2 chunks: h2c = relu(h1 @ W2 + b2); out += h2c @ W3 ----
  v8f acc3a = {0,0,0,0,0,0,0,0};             // rows 0..15 of out, cols owned by this wave
  v8f acc3b = {0,0,0,0,0,0,0,0};             // rows 16..31
  const int ocol = wave * 16 + ln16;         // D_OUT column owned by this wave

  for (int c = 0; c < H2D / H2_CHUNK; ++c) {
#pragma unroll
    for (int t = 0; t < 2; ++t) {
      const int nt2  = wave + (t << 3);      // tile index inside chunk [0,16)
      const int lcol = nt2 * 16 + ln16;      // column within chunk
      const int col  = c * H2_CHUNK + lcol;  // global h2 column
      v8f acc0 = {0,0,0,0,0,0,0,0};
      v8f acc1 = {0,0,0,0,0,0,0,0};
#pragma unroll 4
      for (int k = 0; k < H1D; k += 4) {
        const int k0 = k + kofs;
        v2f a0, a1, bb;
        a0.x = sH1[ln16 * SH1_STRIDE + k0];
        a0.y = sH1[ln16 * SH1_STRIDE + k0 + 1];
        a1.x = sH1[(ln16 + 16) * SH1_STRIDE + k0];
        a1.y = sH1[(ln16 + 16) * SH1_STRIDE + k0 + 1];
        bb.x = W2e[(size_t)k0 * H2D + col];
        bb.y = W2e[(size_t)(k0 + 1) * H2D + col];
        acc0 = wmma4(a0, bb, acc0);
        acc1 = wmma4(a1, bb, acc1);
      }
      const float bias = b2e[col];
#pragma unroll
      for (int r = 0; r < 8; ++r) {
        const int row = r + (hi << 3);
        sH2[row * SH2_STRIDE + lcol]        = fmaxf(acc0[r] + bias, 0.0f);
        sH2[(row + 16) * SH2_STRIDE + lcol] = fmaxf(acc1[r] + bias, 0.0f);
      }
    }
    __syncthreads();

    // Layer-3 partial accumulation from this chunk (K = 256)
#pragma unroll 4
    for (int k = 0; k < H2_CHUNK; k += 4) {
      const int k0 = k + kofs;
      const int gk = c * H2_CHUNK + k0;
      v2f a0, a1, bb;
      a0.x = sH2[ln16 * SH2_STRIDE + k0];
      a0.y = sH2[ln16 * SH2_STRIDE + k0 + 1];
      a1.x = sH2[(ln16 + 16) * SH2_STRIDE + k0];
      a1.y = sH2[(ln16 + 16) * SH2_STRIDE + k0 + 1];
      bb.x = W3e[(size_t)gk * D_OUT + ocol];
      bb.y = W3e[(size_t)(gk + 1) * D_OUT + ocol];
      acc3a = wmma4(a0, bb, acc3a);
      acc3b = wmma4(a1, bb, acc3b);
    }
    __syncthreads();
  }

  // ---- Finalize q/kn = relu(out + b3) into LDS (reuse sX region, stride D_OUT) ----
  float* sOut = sX;
  const float bias3 = b3e[ocol];
#pragma unroll
  for (int r = 0; r < 8; ++r) {
    const int row = r + (hi << 3);
    sOut[row * D_OUT + ocol]        = fmaxf(acc3a[r] + bias3, 0.0f);
    sOut[(row + 16) * D_OUT + ocol] = fmaxf(acc3b[r] + bias3, 0.0f);
  }
  __syncthreads();

  // ---- logits[b][k] = dot(q, kn_k) / T ----
  if (tid < NK) {
    float s = 0.0f;
#pragma unroll 8
    for (int d = 0; d < D_OUT; ++d)
      s += sOut[d] * sOut[(1 + tid) * D_OUT + d];
    out[b * NK + tid] = s * (1.0f / 0.12f);
  }
}

extern "C" void kernel_launch(void* const* d_in, const int* in_sizes, int n_in,
                              void* d_out, int out_size, void* d_ws, size_t ws_size,
                              hipStream_t stream) {
  (void)in_sizes; (void)n_in; (void)out_size; (void)d_ws; (void)ws_size;
  const float* v2s = (const float*)d_in[0];
  const float* W1  = (const float*)d_in[1];
  const float* b1  = (const float*)d_in[2];
  const float* W2  = (const float*)d_in[3];
  const float* b2  = (const float*)d_in[4];
  const float* W3  = (const float*)d_in[5];
  const float* b3  = (const float*)d_in[6];
  const int* tar_atts   = (const int*)d_in[7];
  const int* sample_idx = (const int*)d_in[8];
  float* out = (float*)d_out;

  const size_t lds_bytes =
      (size_t)(32 * SX_STRIDE + 32 * SH1_STRIDE + 32 * SH2_STRIDE) * sizeof(float); // ~204 KB

  simclr_expert_mlp_kernel<<<dim3(NB), dim3(256), lds_bytes, stream>>>(
      v2s, W1, b1, W2, b2, W3, b3, tar_atts, sample_idx, out);
}